// NodeEmbeddingHead_4844723110433
// MI455X (gfx1250) — compile-verified
//
#include <hip/hip_runtime.h>

typedef __attribute__((ext_vector_type(2))) float v2f;
typedef __attribute__((ext_vector_type(8))) float v8f;

#define WAVES_PER_BLOCK 8
#define PTS_PER_WAVE    16
#define PTS_PER_BLOCK   (WAVES_PER_BLOCK * PTS_PER_WAVE)   // 128
#define C_DIM           128
#define N_DIM           64
#define HW              64
#define HWHW            4096
#define TOTAL_PTS       (32 * 4096)

// LDS strides chosen for bank-conflict-free WMMA fragment access:
//  sampled: 132 floats/row  -> A-frag b64 loads hit banks (4m+kb)%64, disjoint
//  h:        68 floats/row  -> same property for the K=64 tile
//  weights: fragment-major float2, K-pair row stride 80 float2 (160 dwords
//           == 32 mod 64 banks) -> lane halves use disjoint 32-bank halves
#define S_STRIDE  132
#define H_STRIDE  68
#define WF_STRIDE 80     // in float2 units

__global__ __launch_bounds__(256)
void node_embed_kernel(const float* __restrict__ fm,
                       const float* __restrict__ coords,
                       const float* __restrict__ W1,
                       const float* __restrict__ b1,
                       const float* __restrict__ W2,
                       const float* __restrict__ b2,
                       float* __restrict__ out)
{
    __shared__ __align__(16) float lds_s[WAVES_PER_BLOCK * PTS_PER_WAVE * S_STRIDE];
    __shared__ __align__(16) float lds_h[WAVES_PER_BLOCK * PTS_PER_WAVE * H_STRIDE];
    // fragment-major: lds_w1f[kp*WF_STRIDE + n] = {W1[2kp][n], W1[2kp+1][n]}
    __shared__ __align__(16) v2f lds_w1f[(C_DIM / 2) * WF_STRIDE];
    __shared__ __align__(16) v2f lds_w2f[(N_DIM / 2) * WF_STRIDE];

    const int tid   = threadIdx.x;
    const int lane  = tid & 31;
    const int wave  = tid >> 5;
    const int laneM = lane & 15;   // matrix row (A) or column (B/C)
    const int laneH = lane >> 4;   // lane half -> K offset of +2

    // ---- Stage 0: preload weights into LDS in B-fragment layout ----
    for (int q = tid; q < (C_DIM / 2) * N_DIM; q += 256) {
        const int kp = q >> 6, n = q & 63;
        v2f w; w[0] = W1[(2 * kp) * N_DIM + n];
               w[1] = W1[(2 * kp + 1) * N_DIM + n];
        lds_w1f[kp * WF_STRIDE + n] = w;
    }
    for (int q = tid; q < (N_DIM / 2) * N_DIM; q += 256) {
        const int kp = q >> 6, n = q & 63;
        v2f w; w[0] = W2[(2 * kp) * N_DIM + n];
               w[1] = W2[(2 * kp + 1) * N_DIM + n];
        lds_w2f[kp * WF_STRIDE + n] = w;
    }

    // ---- Stage 1: bilinear gather, 2 lanes per point, 64 channels per lane ----
    const int wavePt = blockIdx.x * PTS_PER_BLOCK + wave * PTS_PER_WAVE;
    {
        const int   m  = laneM;
        const int   p  = wavePt + m;          // global point id = b*4096 + t
        const int   bb = p >> 12;             // batch index
        const float y  = coords[2 * p + 0];
        const float x  = coords[2 * p + 1];
        const float y0f = floorf(y), x0f = floorf(x);
        const float ty = y - y0f, tx = x - x0f;
        int y0 = (int)y0f; y0 = min(max(y0, 0), HW - 1);
        int x0 = (int)x0f; x0 = min(max(x0, 0), HW - 1);
        const int y1 = min(y0 + 1, HW - 1);
        const int x1 = min(x0 + 1, HW - 1);
        const float w00 = (1.f - ty) * (1.f - tx);
        const float w01 = (1.f - ty) * tx;
        const float w10 = ty * (1.f - tx);
        const float w11 = ty * tx;
        const int o00 = y0 * HW + x0, o01 = y0 * HW + x1;
        const int o10 = y1 * HW + x0, o11 = y1 * HW + x1;

        const float* fmb  = fm + ((size_t)bb * C_DIM + (size_t)laneH * 64) * HWHW;
        float*       sdst = &lds_s[(wave * PTS_PER_WAVE + m) * S_STRIDE + laneH * 64];
        #pragma unroll 8
        for (int j = 0; j < 64; ++j) {
            const float* pc = fmb + (size_t)j * HWHW;
            sdst[j] = w00 * pc[o00] + w01 * pc[o01] + w10 * pc[o10] + w11 * pc[o11];
        }
    }
    __syncthreads();   // weights visible to all waves (gather tile is wave-local)

    // ---- Stage 2: GEMM1  h = relu(S[16x128] @ W1[128x64] + b1) ----
    const float* sbase = &lds_s[wave * PTS_PER_WAVE * S_STRIDE];
    float*       hbase = &lds_h[wave * PTS_PER_WAVE * H_STRIDE];

    v8f acc[4];
    #pragma unroll
    for (int nt = 0; nt < 4; ++nt) acc[nt] = 0.f;

    #pragma unroll 4
    for (int kk = 0; kk < C_DIM / 4; ++kk) {
        const int kb = kk * 4 + laneH * 2;           // this lane-half's K base
        // A fragment: row M=laneM, (K=kb, kb+1) -> one b64 LDS load
        v2f a = *(const v2f*)&sbase[laneM * S_STRIDE + kb];
        const v2f* wrow = &lds_w1f[(2 * kk + laneH) * WF_STRIDE + laneM];
        #pragma unroll
        for (int nt = 0; nt < 4; ++nt) {
            v2f bf = wrow[nt * 16];                  // one contiguous b64 load
            acc[nt] = __builtin_amdgcn_wmma_f32_16x16x4_f32(
                false, a, false, bf, (short)0, acc[nt], false, false);
        }
    }

    // bias + relu, scatter C layout (M = i + 8*laneH, N = nt*16 + laneM) into h tile
    #pragma unroll
    for (int nt = 0; nt < 4; ++nt) {
        const float bias = b1[nt * 16 + laneM];
        #pragma unroll
        for (int i = 0; i < 8; ++i) {
            float v = acc[nt][i] + bias;
            v = fmaxf(v, 0.f);
            hbase[(i + 8 * laneH) * H_STRIDE + nt * 16 + laneM] = v;
        }
    }

    // ---- Stage 3: GEMM2  e = h[16x64] @ W2[64x64] + b2 ----
    v8f acc2[4];
    #pragma unroll
    for (int nt = 0; nt < 4; ++nt) acc2[nt] = 0.f;

    #pragma unroll 4
    for (int kk = 0; kk < N_DIM / 4; ++kk) {
        const int kb = kk * 4 + laneH * 2;
        v2f a = *(const v2f*)&hbase[laneM * H_STRIDE + kb];
        const v2f* wrow = &lds_w2f[(2 * kk + laneH) * WF_STRIDE + laneM];
        #pragma unroll
        for (int nt = 0; nt < 4; ++nt) {
            v2f bf = wrow[nt * 16];
            acc2[nt] = __builtin_amdgcn_wmma_f32_16x16x4_f32(
                false, a, false, bf, (short)0, acc2[nt], false, false);
        }
    }

    // ---- Stage 4: +b2, L2-normalize each point's 64-vector, store ----
    float bb2[4];
    #pragma unroll
    for (int nt = 0; nt < 4; ++nt) bb2[nt] = b2[nt * 16 + laneM];

    #pragma unroll
    for (int i = 0; i < 8; ++i) {
        float v0 = acc2[0][i] + bb2[0];
        float v1 = acc2[1][i] + bb2[1];
        float v2 = acc2[2][i] + bb2[2];
        float v3 = acc2[3][i] + bb2[3];
        float s = v0 * v0 + v1 * v1 + v2 * v2 + v3 * v3;
        // reduce across the 16 lanes holding this point's row (low 4 lane bits)
        s += __shfl_xor(s, 1, 32);
        s += __shfl_xor(s, 2, 32);
        s += __shfl_xor(s, 4, 32);
        s += __shfl_xor(s, 8, 32);
        const float inv = 1.0f / fmaxf(sqrtf(s), 1e-12f);

        const int M    = i + 8 * laneH;
        const int prow = (wavePt + M) * N_DIM + laneM;
        out[prow +  0] = v0 * inv;
        out[prow + 16] = v1 * inv;
        out[prow + 32] = v2 * inv;
        out[prow + 48] = v3 * inv;
    }
}

extern "C" void kernel_launch(void* const* d_in, const int* in_sizes, int n_in,
                              void* d_out, int out_size, void* d_ws, size_t ws_size,
                              hipStream_t stream) {
    const float* fm     = (const float*)d_in[0];
    const float* coords = (const float*)d_in[1];
    const float* W1     = (const float*)d_in[2];
    const float* b1     = (const float*)d_in[3];
    const float* W2     = (const float*)d_in[4];
    const float* b2     = (const float*)d_in[5];
    float* out = (float*)d_out;

    const int blocks = TOTAL_PTS / PTS_PER_BLOCK;   // 1024, exact
    node_embed_kernel<<<blocks, 256, 0, stream>>>(fm, coords, W1, b1, W2, b2, out);
}